// VectorQuantizer_6768868459155
// MI455X (gfx1250) — compile-verified
//
#include <hip/hip_runtime.h>
#include <cstdint>

typedef __attribute__((ext_vector_type(2))) float v2f;
typedef __attribute__((ext_vector_type(8))) float v8f;

#define D_DIM 64
#define K_CODES 512
#define ROWS_PER_WAVE 16
#define WAVES_PER_BLOCK 8
#define ROWS_PER_BLOCK (ROWS_PER_WAVE * WAVES_PER_BLOCK)

__global__ __launch_bounds__(256)
void vq_argmin_wmma_kernel(const float* __restrict__ X,
                           const float* __restrict__ E,
                           float* __restrict__ out) {
    // Codebook resident in LDS: [d][k] row-major, 64*512*4 = 128 KB (of 320 KB/WGP)
    __shared__ float Es[D_DIM * K_CODES];
    __shared__ float eN[K_CODES];            // ||e_k||^2
    __shared__ int   idxs[ROWS_PER_BLOCK];   // per-row argmin result

    const int tid  = threadIdx.x;
    const int lane = tid & 31;
    const int wid  = tid >> 5;

    // ---- Async fill of codebook into LDS (CDNA5 ASYNCcnt path) ----
    // 128 KB / (256 threads * 16 B) = 32 b128 async copies per thread, fully coalesced.
    {
        uint32_t ldsbase = (uint32_t)(uintptr_t)&Es[0];   // low 32 bits of LDS flat addr = LDS offset
        const char* gbase = (const char*)E;
        #pragma unroll
        for (int i = 0; i < 32; ++i) {
            uint32_t off   = (uint32_t)(i * 256 + tid) * 16u;
            uint32_t laddr = ldsbase + off;
            uint64_t gaddr = (uint64_t)(uintptr_t)(gbase + off);
            asm volatile("global_load_async_to_lds_b128 %0, %1, off"
                         :: "v"(laddr), "v"(gaddr) : "memory");
        }
        asm volatile("s_wait_asynccnt 0" ::: "memory");
    }
    __syncthreads();

    // ---- Codebook column norms ||e_k||^2 (argmin objective: ||e||^2 - 2 x.e) ----
    for (int k = tid; k < K_CODES; k += 256) {
        float s = 0.f;
        #pragma unroll 8
        for (int d = 0; d < D_DIM; ++d) { float e = Es[d * K_CODES + k]; s += e * e; }
        eN[k] = s;
    }
    __syncthreads();

    const int row_base = blockIdx.x * ROWS_PER_BLOCK + wid * ROWS_PER_WAVE;

    // ---- A fragments: A = -2*x, WMMA 16x16x4 f32 layout ----
    // lanes 0-15 : M = lane, VGPR0=K0, VGPR1=K1 ; lanes 16-31 : same M, K2/K3
    const int m  = lane & 15;
    const int kh = lane >> 4;     // 0/1 -> K offset 2*kh within each 4-chunk
    v2f a[16];
    {
        const float* xp = X + (size_t)(row_base + m) * D_DIM + (kh << 1);
        #pragma unroll
        for (int c = 0; c < 16; ++c) {
            v2f t = *(const v2f*)(xp + (c << 2));
            a[c] = t * -2.0f;
        }
    }

    float minv[8];
    int   mini[8];
    #pragma unroll
    for (int r = 0; r < 8; ++r) { minv[r] = __builtin_inff(); mini[r] = 0; }

    // ---- Sweep 512 codes as 32 column tiles of 16; D=64 -> 16 accumulated WMMAs/tile ----
    for (int t = 0; t < K_CODES / 16; ++t) {
        const int col = t * 16 + m;
        const float en = eN[col];
        v8f c = { en, en, en, en, en, en, en, en };   // C init = ||e_col||^2
        #pragma unroll
        for (int cc = 0; cc < 16; ++cc) {
            const int d0 = (cc << 2) + (kh << 1);
            v2f b;
            b.x = Es[(d0 + 0) * K_CODES + col];
            b.y = Es[(d0 + 1) * K_CODES + col];
            // D = A*B + C : (-2x).E + ||e||^2
            c = __builtin_amdgcn_wmma_f32_16x16x4_f32(false, a[cc], false, b,
                                                      (short)0, c, false, false);
        }
        // C layout: VGPR r, this lane -> row (r + 8*kh), column `col`
        #pragma unroll
        for (int r = 0; r < 8; ++r) {
            if (c[r] < minv[r]) { minv[r] = c[r]; mini[r] = col; }
        }
    }

    // ---- Argmin across the 16 column lanes of each half (wave32 shuffle tree) ----
    #pragma unroll
    for (int r = 0; r < 8; ++r) {
        float v = minv[r]; int ix = mini[r];
        #pragma unroll
        for (int s = 8; s >= 1; s >>= 1) {
            float ov = __shfl_xor(v, s, 32);
            int   oi = __shfl_xor(ix, s, 32);
            if (ov < v || (ov == v && oi < ix)) { v = ov; ix = oi; }  // first-min tiebreak
        }
        if (m == 0) idxs[wid * ROWS_PER_WAVE + kh * 8 + r] = ix;
    }
    __syncthreads();

    // ---- Gather quantized rows straight from the LDS codebook ----
    {
        const int rrow = lane >> 1;        // 2 lanes per row
        const int half = lane & 1;         // each lane writes 32 contiguous floats
        const int row  = row_base + rrow;
        const int k    = idxs[wid * ROWS_PER_WAVE + rrow];
        float* op = out + (size_t)row * D_DIM + half * 32;
        #pragma unroll
        for (int j = 0; j < 8; ++j) {
            const int d = half * 32 + j * 4;
            float4 q;
            q.x = Es[(d + 0) * K_CODES + k];
            q.y = Es[(d + 1) * K_CODES + k];
            q.z = Es[(d + 2) * K_CODES + k];
            q.w = Es[(d + 3) * K_CODES + k];
            *(float4*)(op + j * 4) = q;
        }
    }
}

extern "C" void kernel_launch(void* const* d_in, const int* in_sizes, int n_in,
                              void* d_out, int out_size, void* d_ws, size_t ws_size,
                              hipStream_t stream) {
    const float* X = (const float*)d_in[0];   // [B,H,W,D] f32, flat N x 64
    const float* E = (const float*)d_in[1];   // [D,K] = [64,512] f32
    float* out = (float*)d_out;               // [B,H,W,D] f32 (== quantized)

    const int nRows  = in_sizes[0] / D_DIM;          // 131072
    const int blocks = nRows / ROWS_PER_BLOCK;       // 1024 (N divisible by 128 here)
    vq_argmin_wmma_kernel<<<blocks, 256, 0, stream>>>(X, E, out);
}